// Atoms_22144851378605
// MI455X (gfx1250) — compile-verified
//
#include <hip/hip_runtime.h>
#include <math.h>

#define NSMP 32768
#define SMASK 32767
#define NFRM 128
#define MDIM 128
#define NHARM 64
#define NBATCH 32
#define PI_F 3.14159265358979f
#define MINF (30.0f/11025.0f)
#define MAXF (3000.0f/11025.0f)
#define INV_SQRT_N 0.005524271728019903f  // 1/sqrt(32768)

typedef float v2f __attribute__((ext_vector_type(2)));
typedef float v8f __attribute__((ext_vector_type(8)));

__device__ __forceinline__ float gumbelf(float u) {
  return -logf(-logf(u + 1e-10f) + 1e-10f);
}
__device__ __forceinline__ float sigmoidf_(float x) { return 1.0f / (1.0f + expf(-x)); }

// linterp(128 -> 32768): frame indices + fractional weight for sample t
__device__ __forceinline__ void frame_of(int t, int& lo, int& hi, float& wf) {
  if (t < 128) { lo = 0; hi = 0; wf = 0.0f; }
  else if (t < 32640) {
    int L = (t - 128) >> 8; int j = (t - 128) & 255;
    lo = L; hi = L + 1; wf = ((float)j + 0.5f) * (1.0f / 256.0f);
  } else { lo = 127; hi = 127; wf = 0.0f; }
}

// closed-form inclusive cumsum of linterp'ed phase increments at sample t
__device__ __forceinline__ float phase_at(int t, const float* __restrict__ inc,
                                          const float* __restrict__ base) {
  if (t < 128) return (float)(t + 1) * inc[0];
  if (t < 32640) {
    int L = (t - 128) >> 8;
    float j1 = (float)(((t - 128) & 255) + 1);
    float iL = inc[L];
    float d  = inc[L + 1] - iL;
    return base[L] + j1 * iL + d * j1 * j1 * (1.0f / 512.0f);
  }
  return base[127] + (float)(t - 32639) * inc[127];
}

// ---------------- K0: small heads (sw gate, f0, f0_var -> per-frame increments, loc shift)
__global__ void k_heads(const float* __restrict__ x,
                        const float* __restrict__ Wsw, const float* __restrict__ bsw,
                        const float* __restrict__ Wf0, const float* __restrict__ bf0,
                        const float* __restrict__ Wfv, const float* __restrict__ bfv,
                        const float* __restrict__ Wloc, const float* __restrict__ bloc,
                        const float* __restrict__ usw, const float* __restrict__ uloc,
                        float* __restrict__ incb, float* __restrict__ swb,
                        int* __restrict__ shiftb) {
  int b = blockIdx.x; int tid = threadIdx.x;
  __shared__ float xs[128];
  __shared__ float red0[128], red1[128], red2[128];
  __shared__ int   ri[128];
  __shared__ float f0s;
  xs[tid] = x[b * 128 + tid];
  __syncthreads();
  float xv = xs[tid];
  float fvdot = 0.f, locdot = 0.f;
  for (int i = 0; i < 128; ++i) {
    float xi = xs[i];
    fvdot  += xi * Wfv[i * 128 + tid];
    locdot += xi * Wloc[i * 128 + tid];
  }
  red0[tid] = xv * Wsw[tid * 2 + 0];
  red1[tid] = xv * Wsw[tid * 2 + 1];
  red2[tid] = xv * Wf0[tid];
  __syncthreads();
  for (int s = 64; s > 0; s >>= 1) {
    if (tid < s) { red0[tid] += red0[tid + s]; red1[tid] += red1[tid + s]; red2[tid] += red2[tid + s]; }
    __syncthreads();
  }
  if (tid == 0) {
    float a0 = red0[0] + bsw[0] + gumbelf(usw[b * 2 + 0]);
    float a1 = red1[0] + bsw[1] + gumbelf(usw[b * 2 + 1]);
    swb[b] = (a0 >= a1) ? 1.0f : 0.0f;   // argmax tie -> index 0
    float f0 = sigmoidf_(red2[0] + bf0[0]);
    f0 = f0 * f0;
    f0s = MINF + f0 * (MAXF - MINF);
  }
  __syncthreads();
  float fv  = sigmoidf_(fvdot + bfv[tid]) * 2.0f - 1.0f;
  float f0t = f0s + fv * f0s * 0.01f;
  for (int h = 0; h < 64; ++h) {
    float v = f0t * (float)(h + 1);
    v = (v > 1.0f) ? 0.0f : v;          // zero aliased harmonics, then *pi
    incb[(size_t)(b * 64 + h) * 128 + tid] = v * PI_F;
  }
  float dv = sigmoidf_(locdot + bloc[tid]) + gumbelf(uloc[b * 128 + tid]);
  red0[tid] = dv; ri[tid] = tid;
  __syncthreads();
  for (int s = 64; s > 0; s >>= 1) {
    if (tid < s) {
      float o = red0[tid + s]; int oi = ri[tid + s];
      if (o > red0[tid] || (o == red0[tid] && oi < ri[tid])) { red0[tid] = o; ri[tid] = oi; }
    }
    __syncthreads();
  }
  if (tid == 0) shiftb[b] = ri[0] * 256;
}

// ---------------- K1: segment-boundary phases + 128-tap noise-filter kernels
__global__ void k_prep(const float* __restrict__ incb, float* __restrict__ baseb,
                       float* __restrict__ nfb) {
  int ch = blockIdx.x * blockDim.x + threadIdx.x;   // 2048 channels
  const float* inc = incb + (size_t)ch * 128;
  float* base = baseb + (size_t)ch * 128;
  float acc = 128.0f * inc[0];
  base[0] = acc;
  for (int L = 1; L < 128; ++L) { acc += 128.0f * (inc[L - 1] + inc[L]); base[L] = acc; }
  float i0 = inc[0];
  float* nf = nfb + (size_t)ch * 128;
  for (int k = 0; k < 128; ++k) {
    float hw = 0.54f - 0.46f * cosf(2.0f * PI_F * (float)k * (1.0f / 128.0f));
    nf[k] = sinf((float)(k + 1) * i0) * hw;
  }
}

// ---------------- K2: noise_u -> [-1,1]
__global__ void k_noise(const float* __restrict__ u, float* __restrict__ np_) {
  int i = blockIdx.x * blockDim.x + threadIdx.x;    // exactly 32*32768
  np_[i] = u[i] * 2.0f - 1.0f;
}

// ---------------- K3: amp = (x @ W_amp + b)^2 via WMMA f32 16x16x4  (M=32,K=128,N=16384)
__global__ void k_amp(const float* __restrict__ x, const float* __restrict__ W,
                      const float* __restrict__ bias, float* __restrict__ ampf) {
  int lane = threadIdx.x & 31;
  int wv   = threadIdx.x >> 5;
  int gw   = blockIdx.x * 8 + wv;       // 2048 wave tiles
  int nt = gw >> 1, mt = gw & 1;
  int m = mt * 16 + (lane & 15);
  int n = nt * 16 + (lane & 15);
  int khalf = (lane >> 4) << 1;         // lanes 0-15 -> K{0,1}; 16-31 -> K{2,3}
  v8f c = {};
  for (int k = 0; k < 32; ++k) {
    int k0 = k * 4 + khalf;
    v2f a, bm;
    a.x  = x[m * 128 + k0];
    a.y  = x[m * 128 + k0 + 1];
    bm.x = W[(size_t)k0 * 16384 + n];
    bm.y = W[(size_t)(k0 + 1) * 16384 + n];
    c = __builtin_amdgcn_wmma_f32_16x16x4_f32(false, a, false, bm, (short)0, c, false, false);
  }
  float bn = bias[n];
  int rbase = mt * 16 + ((lane >> 4) << 3);   // C row = r + 8*(lane>=16)
#pragma unroll
  for (int r = 0; r < 8; ++r) {
    float v = c[r] + bn;
    ampf[(size_t)(rbase + r) * 16384 + n] = v * v;
  }
}

// ---------------- K4: pass 1 of noise conv (implicit GEMM) -> per-channel max|nb|
__global__ void k_convmax(const float* __restrict__ nfb, const float* __restrict__ np_,
                          unsigned* __restrict__ chmax) {
  int lane = threadIdx.x & 31;
  int wv   = threadIdx.x >> 5;
  int gw   = blockIdx.x * 8 + wv;       // 32 b * 4 mt * 2048 tt = 262144 waves
  int tt = gw & 2047;
  int mt = (gw >> 11) & 3;
  int b  = gw >> 13;
  int t = tt * 16 + (lane & 15);
  int m = mt * 16 + (lane & 15);
  int khalf = (lane >> 4) << 1;
  const float* nf = nfb + (size_t)b * 64 * 128;
  const float* nz = np_ + (size_t)b * NSMP;
  v8f c = {};
  for (int k = 0; k < 32; ++k) {
    int k0 = k * 4 + khalf;
    v2f a, bm;
    a.x  = nf[m * 128 + k0];
    a.y  = nf[m * 128 + k0 + 1];
    bm.x = nz[(t - k0) & SMASK];        // circular conv column
    bm.y = nz[(t - k0 - 1) & SMASK];
    c = __builtin_amdgcn_wmma_f32_16x16x4_f32(false, a, false, bm, (short)0, c, false, false);
  }
#pragma unroll
  for (int r = 0; r < 8; ++r) {
    float v = fabsf(c[r]) * INV_SQRT_N;
    for (int off = 1; off < 16; off <<= 1) v = fmaxf(v, __shfl_xor(v, off));
    if ((lane & 15) == 0) {
      int ch = b * 64 + mt * 16 + r + ((lane >> 4) << 3);
      atomicMax(&chmax[ch], __float_as_uint(v));   // v >= 0: uint order == float order
    }
  }
}

// ---------------- K5: fused final: re-conv noise, normalize, amp-interp, harm sin,
//                       channel-sum, sw gate, circular shift, accumulate to output
__global__ void k_final(const float* __restrict__ nfb, const float* __restrict__ np_,
                        const float* __restrict__ ampf, const float* __restrict__ incb,
                        const float* __restrict__ baseb, const unsigned* __restrict__ chmax,
                        const float* __restrict__ swb, const int* __restrict__ shiftb,
                        float* __restrict__ res) {
  int lane = threadIdx.x & 31;
  int wv   = threadIdx.x >> 5;
  int gw   = blockIdx.x * 8 + wv;
  int tt = gw & 2047;
  int mt = (gw >> 11) & 3;
  int b  = gw >> 13;
  int t = tt * 16 + (lane & 15);
  int khalf = (lane >> 4) << 1;
  int m0 = mt * 16 + (lane & 15);
  const float* nf = nfb + (size_t)b * 64 * 128;
  const float* nz = np_ + (size_t)b * NSMP;
  v8f c = {};
  for (int k = 0; k < 32; ++k) {
    int k0 = k * 4 + khalf;
    v2f a, bm;
    a.x  = nf[m0 * 128 + k0];
    a.y  = nf[m0 * 128 + k0 + 1];
    bm.x = nz[(t - k0) & SMASK];
    bm.y = nz[(t - k0 - 1) & SMASK];
    c = __builtin_amdgcn_wmma_f32_16x16x4_f32(false, a, false, bm, (short)0, c, false, false);
  }
  int flo, fhi; float wf;
  frame_of(t, flo, fhi, wf);
  const float* ampb = ampf + (size_t)b * 16384;
  int halfofs = (lane >> 4) << 3;
  float acc = 0.0f;
#pragma unroll
  for (int r = 0; r < 8; ++r) {                   // noise channels 64..127
    int m  = mt * 16 + r + halfofs;
    int ch = b * 64 + m;
    float mx  = __uint_as_float(chmax[ch]);
    float fac = INV_SQRT_N / (mx + 1e-8f);        // max_norm of scaled conv
    const float* ac = ampb + (size_t)(64 + m) * 128;
    float a0 = ac[flo], a1 = ac[fhi];
    acc += c[r] * fac * (a0 + (a1 - a0) * wf);
  }
  if (mt == 0) {                                  // harmonic channels 0..63 (wave-uniform)
    int hbase = (lane >> 4) * 32;
    for (int hh = 0; hh < 32; ++hh) {
      int h  = hbase + hh;
      int ch = b * 64 + h;
      float p = phase_at(t, incb + (size_t)ch * 128, baseb + (size_t)ch * 128);
      const float* ac = ampb + (size_t)h * 128;
      float a0 = ac[flo], a1 = ac[fhi];
      acc += sinf(p) * (a0 + (a1 - a0) * wf);
    }
  }
  acc += __shfl_xor(acc, 16);                     // combine lane halves (same t)
  if (lane < 16) {
    float sw = swb[b];
    int sh   = shiftb[b];
    atomicAdd(&res[(size_t)b * NSMP + ((t + sh) & SMASK)], sw * acc);
  }
}

extern "C" void kernel_launch(void* const* d_in, const int* in_sizes, int n_in,
                              void* d_out, int out_size, void* d_ws, size_t ws_size,
                              hipStream_t stream) {
  const float* x    = (const float*)d_in[0];
  const float* Wsw  = (const float*)d_in[1];
  const float* bsw  = (const float*)d_in[2];
  const float* Wf0  = (const float*)d_in[3];
  const float* bf0  = (const float*)d_in[4];
  const float* Wfv  = (const float*)d_in[5];
  const float* bfv  = (const float*)d_in[6];
  const float* Wamp = (const float*)d_in[7];
  const float* bamp = (const float*)d_in[8];
  const float* Wloc = (const float*)d_in[9];
  const float* bloc = (const float*)d_in[10];
  const float* usw  = (const float*)d_in[11];
  const float* uloc = (const float*)d_in[12];
  const float* nu   = (const float*)d_in[13];

  float* ws = (float*)d_ws;
  float* incb   = ws;                       // 2048*128
  float* baseb  = incb + 262144;            // 2048*128
  float* nfb    = baseb + 262144;           // 2048*128
  float* ampf   = nfb + 262144;             // 32*16384
  float* noisep = ampf + 524288;            // 32*32768
  float* swb    = noisep + 1048576;         // 32
  unsigned* chmax = (unsigned*)(swb + 32);  // 2048
  int* shiftb   = (int*)(chmax + 2048);     // 32
  float* res    = (float*)d_out;            // 32*32768

  hipMemsetAsync(res, 0, (size_t)NBATCH * NSMP * sizeof(float), stream);
  hipMemsetAsync(chmax, 0, 2048 * sizeof(unsigned), stream);

  k_heads<<<32, 128, 0, stream>>>(x, Wsw, bsw, Wf0, bf0, Wfv, bfv, Wloc, bloc,
                                  usw, uloc, incb, swb, shiftb);
  k_noise<<<4096, 256, 0, stream>>>(nu, noisep);
  k_prep<<<8, 256, 0, stream>>>(incb, baseb, nfb);
  k_amp<<<256, 256, 0, stream>>>(x, Wamp, bamp, ampf);
  k_convmax<<<32768, 256, 0, stream>>>(nfb, noisep, chmax);
  k_final<<<32768, 256, 0, stream>>>(nfb, noisep, ampf, incb, baseb, chmax,
                                     swb, shiftb, res);
}